// NearestNeighborModule_21131239096553
// MI455X (gfx1250) — compile-verified
//
#include <hip/hip_runtime.h>
#include <hip/hip_bf16.h>

typedef __attribute__((ext_vector_type(2))) float v2f;
typedef __attribute__((ext_vector_type(8))) float v8f;

#define NN_B     4096   // queries
#define NN_N     65536  // reference points
#define NN_CHUNK 4096   // points staged per block (16 KB LDS)
#define NN_QW    32     // queries per wave (two 16-wide WMMA B matrices)

// ---------------------------------------------------------------------------
// Init workspace: ws[q] = u64 max so atomic-min works from a clean slate each
// call (harness replays the graph; must be deterministic).
// ---------------------------------------------------------------------------
__global__ __launch_bounds__(256) void nn_init_kernel(unsigned long long* __restrict__ ws) {
    int q = blockIdx.x * 256 + threadIdx.x;
    if (q < NN_B) ws[q] = ~0ull;
}

// ---------------------------------------------------------------------------
// Main kernel: WMMA-based score matrix  S[m][n] = y_m^2 - 2*x_n*y_m
//   A (16x4 fp32): A[m] = [y_m^2, y_m, 0, 0]   (points on M, from LDS)
//   B (4x16 fp32): B[n] = [1, -2*x_n, 0, 0]    (queries on N, loop-invariant)
// D layout: lane L, VGPR r -> D[M = r + 8*(L>=16)][N = L&15]  => lane-local
// argmin + one SWAPX16 ds_swizzle merge per query group.
// Each wave carries TWO query groups (B0, B1) and shares one A build per two
// WMMAs. LDS is filled with the CDNA5 async copy path (ASYNCcnt-tracked).
// ---------------------------------------------------------------------------
__global__ __launch_bounds__(256) void nn_main_kernel(const float* __restrict__ x,
                                                      const float* __restrict__ y,
                                                      unsigned long long* __restrict__ ws) {
    __shared__ float s_y[NN_CHUNK];

    const int tid       = threadIdx.x;
    const int qgrp      = blockIdx.x;              // 0..127 (32 queries each)
    const int chunkBase = blockIdx.y * NN_CHUNK;   // 0, 4096, ...

    // --- Async stage of this block's 16 KB point chunk into LDS ------------
    {
        // Low 32 bits of a generic pointer to LDS == LDS byte offset.
        const unsigned ldsBase = (unsigned)(uintptr_t)(void*)s_y;
        const float4*  gsrc    = (const float4*)(y + chunkBase);
#pragma unroll
        for (int i = 0; i < NN_CHUNK / 4 / 256; ++i) {   // 4 issues x 4 KB
            unsigned      ldst = ldsBase + (unsigned)((tid + i * 256) * 16);
            const float4* gptr = gsrc + tid + i * 256;
            asm volatile("global_load_async_to_lds_b128 %0, %1, off"
                         :: "v"(ldst), "v"(gptr)
                         : "memory");
        }
        asm volatile("s_wait_asynccnt 0x0" ::: "memory");
    }
    __syncthreads();

    const int lane   = tid & 31;         // wave32
    const int wave   = tid >> 5;         // 0..7
    const bool lo    = lane < 16;
    const int qq     = lane & 15;
    const int q0     = qgrp * NN_QW + qq;        // query group 0
    const int q1     = q0 + 16;                  // query group 1
    const int rowOff = (lane >> 4) << 3;         // 0 or 8

    // Two loop-invariant B matrices (one per query group). Upper lanes carry
    // B's K=2,3 rows which must be zero.
    const float xq0 = x[q0];
    const float xq1 = x[q1];
    v2f bmat0, bmat1;
    bmat0.x = lo ? 1.0f : 0.0f;
    bmat0.y = lo ? (-2.0f * xq0) : 0.0f;
    bmat1.x = bmat0.x;
    bmat1.y = lo ? (-2.0f * xq1) : 0.0f;

    const v8f czero = {};

    float best0 = 3.4028234663852886e38f, best1 = 3.4028234663852886e38f;
    int   idx0  = 0, idx1 = 0;

    // Tile-min tree (lowers to v_min3_num_f32) + rare argmin recovery.
    auto track = [&](const v8f& d, float& best, int& bestIdx, int ibase) {
        float m0 = fminf(d[0], d[1]);
        float m1 = fminf(d[2], d[3]);
        float m2 = fminf(d[4], d[5]);
        float m3 = fminf(d[6], d[7]);
        float n0 = fminf(m0, m1);
        float n1 = fminf(m2, m3);
        float tmin = fminf(n0, n1);
        if (tmin < best) {                 // strict <  => earliest tile wins
            best = tmin;
            int ri = 7;
#pragma unroll
            for (int r = 6; r >= 0; --r)   // smallest matching r wins ties
                ri = (d[r] == tmin) ? r : ri;
            bestIdx = ibase + ri;
        }
    };

    const int wbase = wave * (NN_CHUNK / 8);     // 512 points per wave

#pragma unroll 1
    for (int g = 0; g < (NN_CHUNK / 8) / 64; ++g) {   // 8 groups of 4 tiles
        const int gbase = wbase + g * 64;

        // Straight-line preload of 4 tiles' A inputs: ds_loads batch into one
        // issue group (64 B apart -> ds_load_2addr candidates), single wait.
        float yl[4];
#pragma unroll
        for (int u = 0; u < 4; ++u)
            yl[u] = s_y[gbase + u * 16 + qq];

#pragma unroll
        for (int u = 0; u < 4; ++u) {
            const float yv = lo ? yl[u] : 0.0f;   // upper lanes: A's K=2,3 = 0
            v2f amat;
            amat.x = yv * yv;
            amat.y = yv;

            // Two independent WMMAs share one A build (no D->A/B hazard).
            v8f d0 = __builtin_amdgcn_wmma_f32_16x16x4_f32(
                false, amat, false, bmat0, (short)0, czero, false, false);
            v8f d1 = __builtin_amdgcn_wmma_f32_16x16x4_f32(
                false, amat, false, bmat1, (short)0, czero, false, false);

            const int ibase = chunkBase + gbase + u * 16 + rowOff;
            track(d0, best0, idx0, ibase);
            track(d1, best1, idx1, ibase);
        }
    }

    // Merge lane L with lane L+16 (rows 8..15 of each tile) via SWAPX16.
    int   oI0 = __builtin_amdgcn_ds_swizzle(idx0, 0x401F);
    int   oS0 = __builtin_amdgcn_ds_swizzle(__float_as_int(best0), 0x401F);
    int   oI1 = __builtin_amdgcn_ds_swizzle(idx1, 0x401F);
    int   oS1 = __builtin_amdgcn_ds_swizzle(__float_as_int(best1), 0x401F);
    float oB0 = __int_as_float(oS0);
    float oB1 = __int_as_float(oS1);

    if (lo) {
        if (oB0 < best0 || (oB0 == best0 && oI0 < idx0)) { best0 = oB0; idx0 = oI0; }
        if (oB1 < best1 || (oB1 == best1 && oI1 < idx1)) { best1 = oB1; idx1 = oI1; }

        // Pack (score, idx) into a sortable u64 key; atomic min gives the
        // globally smallest score with smallest index on ties (== jnp.argmin).
        unsigned u0 = __float_as_uint(best0);
        u0 = (u0 & 0x80000000u) ? ~u0 : (u0 | 0x80000000u);
        unsigned u1 = __float_as_uint(best1);
        u1 = (u1 & 0x80000000u) ? ~u1 : (u1 | 0x80000000u);

        atomicMin(&ws[q0], ((unsigned long long)u0 << 32) | (unsigned)idx0);
        atomicMin(&ws[q1], ((unsigned long long)u1 << 32) | (unsigned)idx1);
    }
}

// ---------------------------------------------------------------------------
// Finalize: gather accuracy at the winning index.
// ---------------------------------------------------------------------------
__global__ __launch_bounds__(256) void nn_final_kernel(const unsigned long long* __restrict__ ws,
                                                       const float* __restrict__ acc,
                                                       float* __restrict__ out) {
    int q = blockIdx.x * 256 + threadIdx.x;
    if (q < NN_B) {
        unsigned idx = (unsigned)ws[q];   // low 32 bits = argmin index
        out[q] = acc[idx];
    }
}

extern "C" void kernel_launch(void* const* d_in, const int* in_sizes, int n_in,
                              void* d_out, int out_size, void* d_ws, size_t ws_size,
                              hipStream_t stream) {
    const float* x   = (const float*)d_in[0];   // (4096,)  queries
    const float* y   = (const float*)d_in[1];   // (65536,) reference points
    const float* acc = (const float*)d_in[2];   // (65536,) accuracy values
    float* out = (float*)d_out;                 // (4096,) fp32
    unsigned long long* ws = (unsigned long long*)d_ws;   // 4096 * 8 = 32 KB

    (void)in_sizes; (void)n_in; (void)out_size; (void)ws_size;

    nn_init_kernel<<<NN_B / 256, 256, 0, stream>>>(ws);

    dim3 grid(NN_B / NN_QW, NN_N / NN_CHUNK);   // (128, 16)
    nn_main_kernel<<<grid, 256, 0, stream>>>(x, y, ws);

    nn_final_kernel<<<NN_B / 256, 256, 0, stream>>>(ws, acc, out);
}